// FlowMatchingLoss_58428735095151
// MI455X (gfx1250) — compile-verified
//
#include <hip/hip_runtime.h>
#include <math.h>

// ---------------------------------------------------------------------------
// FlowMatching OT loss for MI455X (gfx1250).
// Sinkhorn dominates: 50 iters x 2 passes x B*M*K = 6.7G element visits.
// The 256MB cost matrix is never materialized: each 16x16 tile of
//   C_ij = max(|x_i|^2 - 2 x_i.y_j + |y_j|^2, 0)
// is rebuilt by one V_WMMA_F32_16X16X4_F32 with A = [-2x | |x|^2], B = [y ; 1].
// Operands are pre-packed so every lane's A/B fragment is a single branchless
// global_load_b64; tile loops are 2x unrolled so two independent WMMA->VALU
// chains overlap (hides the WMMA data hazard and load latency).
// ---------------------------------------------------------------------------

typedef float v2f __attribute__((ext_vector_type(2)));
typedef float v4f __attribute__((ext_vector_type(4)));
typedef float v8f __attribute__((ext_vector_type(8)));

#define B_      4
#define M_      4096
#define K_      4096
#define REG_    0.1f
#define NITER_  50
#define THRESH_ 1e-5f
#define LOGN_   8.317766166719343f   /* log(4096) */

// workspace layout in float units (all offsets 16B-aligned)
#define OFF_PA     0                          /* B*M*4 : (-2x,-2y,-2z,|x|^2) */
#define OFF_PB     (OFF_PA + B_*M_*4)         /* B*K*4 : ( y0, y1, y2, 1.0 ) */
#define OFF_GS     (OFF_PB + B_*K_*4)         /* B*K*2 : ( g_j , |y_j|^2 )   */
#define OFF_F      (OFF_GS + B_*K_*2)         /* B*M                          */
#define OFF_ROWMAX (OFF_F + B_*M_)            /* B*M                          */
#define OFF_ROWIDX (OFF_ROWMAX + B_*M_)       /* B*M ints                     */
#define OFF_MAXC   (OFF_ROWIDX + B_*M_)       /* B uints, padded to 16        */
#define OFF_ACC    (OFF_MAXC + 16)            /* 3 floats                     */

// ---- WMMA tile: D = A(16x4) * B(4x16), f32 ---------------------------------
__device__ __forceinline__ v8f tile_dots(v2f a, v2f b) {
    v8f c = {};
    // (neg_a, A, neg_b, B, c_mod, C, reuse_a, reuse_b)
    return __builtin_amdgcn_wmma_f32_16x16x4_f32(false, a, false, b,
                                                 (short)0, c, false, false);
}

// A 16x4: lane L holds row M=L&15, K-pair {0,1} (lanes<16) or {2,3} (lanes>=16)
__device__ __forceinline__ v2f frag_a(const float* pA, int i, int hh) {
    return *(const v2f*)(pA + (size_t)i * 4 + hh * 2);
}
// B 4x16: lane L holds col N=L&15, K-pair {0,1} / {2,3}
__device__ __forceinline__ v2f frag_b(const float* pB, int i, int hh) {
    return *(const v2f*)(pB + (size_t)i * 4 + hh * 2);
}

// ---- prep: pack operands, init g/maxc/acc ----------------------------------
__global__ void fm_prep_kernel(const float* x0, const float* xgt,
                               float* pA, float* pB, float* gs,
                               unsigned* maxc, float* acc) {
    int gid = blockIdx.x * blockDim.x + threadIdx.x;
    if (gid < B_ * M_) {
        const float* p = x0 + (size_t)gid * 3;
        float a0 = p[0], a1 = p[1], a2 = p[2];
        v4f va = { -2.0f * a0, -2.0f * a1, -2.0f * a2, a0*a0 + a1*a1 + a2*a2 };
        *(v4f*)(pA + (size_t)gid * 4) = va;
    }
    if (gid < B_ * K_) {
        const float* p = xgt + (size_t)gid * 3;
        float b0 = p[0], b1 = p[1], b2 = p[2];
        v4f vb = { b0, b1, b2, 1.0f };
        *(v4f*)(pB + (size_t)gid * 4) = vb;
        v2f vg = { 0.0f, b0*b0 + b1*b1 + b2*b2 };   // (g=0, |y|^2)
        *(v2f*)(gs + (size_t)gid * 2) = vg;
    }
    if (gid < B_) maxc[gid] = 0u;
    if (gid < 3)  acc[gid] = 0.0f;
}

// ---- per-batch max of C (C >= 0 so uint-bit atomicMax is order-preserving) -
__global__ void fm_maxc_kernel(const float* pA, const float* pB,
                               const float* gs, unsigned* maxc) {
    const int b       = blockIdx.x >> 8;
    const int rowbase = (blockIdx.x & 255) * 16;
    const int lane = threadIdx.x & 31, wave = threadIdx.x >> 5;
    const int hh = lane >> 4, lidx = lane & 15;

    v2f a = frag_a(pA, b * M_ + rowbase + lidx, hh);
    float mx0 = 0.0f, mx1 = 0.0f;
    for (int ct = wave; ct < K_ / 16; ct += 8) {     // 2 tiles / iteration
        int j0 = b * K_ + ct * 16 + lidx;
        int j1 = j0 + 4 * 16;
        v2f bm0 = frag_b(pB, j0, hh);
        v2f bm1 = frag_b(pB, j1, hh);
        float sq0 = gs[(size_t)j0 * 2 + 1];
        float sq1 = gs[(size_t)j1 * 2 + 1];
        v8f d0 = tile_dots(a, bm0);
        v8f d1 = tile_dots(a, bm1);
#pragma unroll
        for (int r = 0; r < 8; ++r) {
            mx0 = fmaxf(mx0, d0[r] + sq0);
            mx1 = fmaxf(mx1, d1[r] + sq1);
        }
    }
    float mx = fmaxf(mx0, mx1);
#pragma unroll
    for (int msk = 16; msk >= 1; msk >>= 1)
        mx = fmaxf(mx, __shfl_xor(mx, msk, 32));
    __shared__ unsigned smx;
    if (threadIdx.x == 0) smx = 0u;
    __syncthreads();
    if (lane == 0) atomicMax(&smx, __float_as_uint(fmaxf(mx, 0.0f)));
    __syncthreads();
    if (threadIdx.x == 0) atomicMax(&maxc[b], smx);
}

// ---- Sinkhorn row pass: f_i = -log(M) - log(sum_j exp(lK_ij + g_j)) --------
// lK+g stays in a narrow negative band for this data; unshifted fp32 exp-sum
// is safe (no overflow; underflow only flushes negligible terms).
__global__ void fm_row_kernel(const float* pA, const float* pB,
                              const float* gs, const unsigned* maxcb,
                              float* f) {
    const int b       = blockIdx.x >> 8;
    const int rowbase = (blockIdx.x & 255) * 16;
    const int lane = threadIdx.x & 31, wave = threadIdx.x >> 5;
    const int hh = lane >> 4, lidx = lane & 15;

    const float scale = -1.0f / (REG_ * fmaxf(__uint_as_float(maxcb[b]), 1e-20f));
    v2f a = frag_a(pA, b * M_ + rowbase + lidx, hh);

    float s[8];
#pragma unroll
    for (int r = 0; r < 8; ++r) s[r] = 0.0f;

    for (int ct = wave; ct < K_ / 16; ct += 8) {     // 2 tiles / iteration
        int j0 = b * K_ + ct * 16 + lidx;
        int j1 = j0 + 4 * 16;
        v2f bm0 = frag_b(pB, j0, hh);
        v2f bm1 = frag_b(pB, j1, hh);
        v2f gs0 = *(const v2f*)(gs + (size_t)j0 * 2);   // (g_j, |y_j|^2)
        v2f gs1 = *(const v2f*)(gs + (size_t)j1 * 2);
        v8f d0 = tile_dots(a, bm0);
        v8f d1 = tile_dots(a, bm1);
#pragma unroll
        for (int r = 0; r < 8; ++r) {
            float c0 = fmaxf(d0[r] + gs0.y, 0.0f);
            float c1 = fmaxf(d1[r] + gs1.y, 0.0f);
            s[r] += __expf(fmaf(c0, scale, gs0.x));
            s[r] += __expf(fmaf(c1, scale, gs1.x));
        }
    }
    // lanes in a 16-lane half share rows {0..7} ({8..15}): reduce over columns
#pragma unroll
    for (int r = 0; r < 8; ++r) {
        float v = s[r];
        v += __shfl_xor(v, 1, 32);
        v += __shfl_xor(v, 2, 32);
        v += __shfl_xor(v, 4, 32);
        v += __shfl_xor(v, 8, 32);
        s[r] = v;
    }
    __shared__ float lds[16];
    if (threadIdx.x < 16) lds[threadIdx.x] = 0.0f;
    __syncthreads();
    if (lidx == 0) {
#pragma unroll
        for (int r = 0; r < 8; ++r) atomicAdd(&lds[r + hh * 8], s[r]);
    }
    __syncthreads();
    if (threadIdx.x < 16)
        f[b * M_ + rowbase + threadIdx.x] = -LOGN_ - __logf(lds[threadIdx.x]);
}

// ---- Sinkhorn col pass: g_j = -log(K) - log(sum_i exp(lK_ij + f_i)) --------
__global__ void fm_col_kernel(const float* pA, const float* pB,
                              const unsigned* maxcb, const float* f, float* gs) {
    const int b       = blockIdx.x >> 8;
    const int colbase = (blockIdx.x & 255) * 16;
    const int lane = threadIdx.x & 31, wave = threadIdx.x >> 5;
    const int hh = lane >> 4, lidx = lane & 15;

    const float scale = -1.0f / (REG_ * fmaxf(__uint_as_float(maxcb[b]), 1e-20f));
    const int j = b * K_ + colbase + lidx;
    v2f bm = frag_b(pB, j, hh);
    const float sqgj = gs[(size_t)j * 2 + 1];

    float cs0 = 0.0f, cs1 = 0.0f;
    for (int rt = wave; rt < M_ / 16; rt += 8) {     // 2 tiles / iteration
        int m0 = b * M_ + rt * 16;
        int m1 = m0 + 4 * 16;
        v2f a0 = frag_a(pA, m0 + lidx, hh);
        v2f a1 = frag_a(pA, m1 + lidx, hh);
        // this lane's 8 D elements are rows m + hh*8 + {0..7}, same col j
        const float* fr0 = f + m0 + hh * 8;
        const float* fr1 = f + m1 + hh * 8;
        v4f fa0 = *(const v4f*)fr0, fa1 = *(const v4f*)(fr0 + 4);
        v4f fb0 = *(const v4f*)fr1, fb1 = *(const v4f*)(fr1 + 4);
        v8f d0 = tile_dots(a0, bm);
        v8f d1 = tile_dots(a1, bm);
        float fv0[8] = { fa0.x, fa0.y, fa0.z, fa0.w, fa1.x, fa1.y, fa1.z, fa1.w };
        float fv1[8] = { fb0.x, fb0.y, fb0.z, fb0.w, fb1.x, fb1.y, fb1.z, fb1.w };
#pragma unroll
        for (int r = 0; r < 8; ++r) {
            float c0 = fmaxf(d0[r] + sqgj, 0.0f);
            float c1 = fmaxf(d1[r] + sqgj, 0.0f);
            cs0 += __expf(fmaf(c0, scale, fv0[r]));
            cs1 += __expf(fmaf(c1, scale, fv1[r]));
        }
    }
    float cs = cs0 + cs1;
    cs += __shfl_xor(cs, 16, 32);   // merge the two row halves (same column)
    __shared__ float lds[16];
    if (threadIdx.x < 16) lds[threadIdx.x] = 0.0f;
    __syncthreads();
    if (lane < 16) atomicAdd(&lds[lidx], cs);
    __syncthreads();
    if (threadIdx.x < 16) {
        int jj = b * K_ + colbase + threadIdx.x;
        gs[(size_t)jj * 2] = -LOGN_ - __logf(lds[threadIdx.x]);
    }
}

// ---- row argmax of log_pi = f_i + lK_ij + g_j ------------------------------
__global__ void fm_argmax_kernel(const float* pA, const float* pB,
                                 const float* gs, const unsigned* maxcb,
                                 const float* f, float* rowmax, int* rowidx) {
    const int b       = blockIdx.x >> 8;
    const int rowbase = (blockIdx.x & 255) * 16;
    const int lane = threadIdx.x & 31, wave = threadIdx.x >> 5;
    const int hh = lane >> 4, lidx = lane & 15;

    const float scale = -1.0f / (REG_ * fmaxf(__uint_as_float(maxcb[b]), 1e-20f));
    v2f a = frag_a(pA, b * M_ + rowbase + lidx, hh);

    float bv[8]; int bj[8];
#pragma unroll
    for (int r = 0; r < 8; ++r) { bv[r] = -3.4e38f; bj[r] = 0; }

    for (int ct = wave; ct < K_ / 16; ct += 8) {     // 2 tiles / iteration
        int jc0 = ct * 16 + lidx;
        int jc1 = jc0 + 4 * 16;
        int j0  = b * K_ + jc0;
        int j1  = b * K_ + jc1;
        v2f bm0 = frag_b(pB, j0, hh);
        v2f bm1 = frag_b(pB, j1, hh);
        v2f gs0 = *(const v2f*)(gs + (size_t)j0 * 2);
        v2f gs1 = *(const v2f*)(gs + (size_t)j1 * 2);
        v8f d0 = tile_dots(a, bm0);
        v8f d1 = tile_dots(a, bm1);
#pragma unroll
        for (int r = 0; r < 8; ++r) {
            float c0 = fmaxf(d0[r] + gs0.y, 0.0f);
            float z0 = fmaf(c0, scale, gs0.x);     // f is constant per row
            if (z0 > bv[r]) { bv[r] = z0; bj[r] = jc0; }
            float c1 = fmaxf(d1[r] + gs1.y, 0.0f);
            float z1 = fmaf(c1, scale, gs1.x);
            if (z1 > bv[r] || (z1 == bv[r] && jc1 < bj[r])) { bv[r] = z1; bj[r] = jc1; }
        }
    }
#pragma unroll
    for (int msk = 1; msk <= 8; msk <<= 1) {
#pragma unroll
        for (int r = 0; r < 8; ++r) {
            float ov = __shfl_xor(bv[r], msk, 32);
            int   oj = __shfl_xor(bj[r], msk, 32);
            if (ov > bv[r] || (ov == bv[r] && oj < bj[r])) { bv[r] = ov; bj[r] = oj; }
        }
    }
    __shared__ float lv[16];
    __shared__ int   lj[16];
    if (threadIdx.x < 16) { lv[threadIdx.x] = -3.4e38f; lj[threadIdx.x] = 0; }
    __syncthreads();
    for (int w = 0; w < 4; ++w) {      // serialized cross-wave merge
        if (wave == w && lidx == 0) {
#pragma unroll
            for (int r = 0; r < 8; ++r) {
                int row = r + hh * 8;
                if (bv[r] > lv[row] || (bv[r] == lv[row] && bj[r] < lj[row])) {
                    lv[row] = bv[r]; lj[row] = bj[r];
                }
            }
        }
        __syncthreads();
    }
    if (threadIdx.x < 16) {
        int gi = b * M_ + rowbase + threadIdx.x;
        rowmax[gi] = f[gi] + lv[threadIdx.x];
        rowidx[gi] = lj[threadIdx.x];
    }
}

// ---- loss partial sums -----------------------------------------------------
__global__ void fm_loss_kernel(const float* x0, const float* xgt,
                               const float* vpred, const float* alpha,
                               const float* rowmax, const int* rowidx,
                               float* acc) {
    __shared__ float sacc[3];
    if (threadIdx.x < 3) sacc[threadIdx.x] = 0.0f;
    __syncthreads();
    int gid = blockIdx.x * blockDim.x + threadIdx.x;
    float vel = 0.0f, sv = 0.0f, bce = 0.0f;
    if (gid < B_ * M_) {
        int b = gid >> 12;                         // / 4096
        float surv = (__expf(rowmax[gid]) > THRESH_) ? 1.0f : 0.0f;
        int idx = rowidx[gid];
        const float* xg = xgt + ((size_t)b * K_ + idx) * 3;
        const float* xp = x0 + (size_t)gid * 3;
        const float* vp = vpred + (size_t)gid * 3;
        float d0 = vp[0] - (xg[0] - xp[0]);
        float d1 = vp[1] - (xg[1] - xp[1]);
        float d2 = vp[2] - (xg[2] - xp[2]);
        vel = surv * (d0*d0 + d1*d1 + d2*d2);
        sv  = surv;
        float z = alpha[gid];
        bce = fmaxf(z, 0.0f) + log1pf(__expf(-fabsf(z))) - z * surv;
    }
#pragma unroll
    for (int msk = 16; msk >= 1; msk >>= 1) {
        vel += __shfl_xor(vel, msk, 32);
        sv  += __shfl_xor(sv,  msk, 32);
        bce += __shfl_xor(bce, msk, 32);
    }
    if ((threadIdx.x & 31) == 0) {
        atomicAdd(&sacc[0], vel);
        atomicAdd(&sacc[1], sv);
        atomicAdd(&sacc[2], bce);
    }
    __syncthreads();
    if (threadIdx.x == 0) {
        atomicAdd(&acc[0], sacc[0]);
        atomicAdd(&acc[1], sacc[1]);
        atomicAdd(&acc[2], sacc[2]);
    }
}

__global__ void fm_finalize_kernel(const float* acc, float* out) {
    if (threadIdx.x == 0 && blockIdx.x == 0) {
        float ns = fmaxf(acc[1], 1.0f);
        out[0] = acc[0] / ns + acc[2] * (1.0f / (float)(B_ * M_));
    }
}

// ---------------------------------------------------------------------------
extern "C" void kernel_launch(void* const* d_in, const int* in_sizes, int n_in,
                              void* d_out, int out_size, void* d_ws, size_t ws_size,
                              hipStream_t stream) {
    const float* x0    = (const float*)d_in[0];
    const float* xgt   = (const float*)d_in[1];
    const float* vpred = (const float*)d_in[2];
    const float* alpha = (const float*)d_in[3];
    // d_in[4] = t : unused by the reference loss

    float*    ws     = (float*)d_ws;
    float*    pA     = ws + OFF_PA;
    float*    pB     = ws + OFF_PB;
    float*    gs     = ws + OFF_GS;
    float*    fbuf   = ws + OFF_F;
    float*    rowmax = ws + OFF_ROWMAX;
    int*      rowidx = (int*)(ws + OFF_ROWIDX);
    unsigned* maxc   = (unsigned*)(ws + OFF_MAXC);
    float*    acc    = ws + OFF_ACC;
    float*    out    = (float*)d_out;

    const dim3 tileGrid(B_ * (M_ / 16));   // 1024 workgroups
    const dim3 blk128(128);                // 4 wave32s per workgroup

    fm_prep_kernel<<<(B_ * M_ + 255) / 256, 256, 0, stream>>>(x0, xgt, pA, pB,
                                                              gs, maxc, acc);
    fm_maxc_kernel<<<tileGrid, blk128, 0, stream>>>(pA, pB, gs, maxc);

    for (int it = 0; it < NITER_; ++it) {
        fm_row_kernel<<<tileGrid, blk128, 0, stream>>>(pA, pB, gs, maxc, fbuf);
        fm_col_kernel<<<tileGrid, blk128, 0, stream>>>(pA, pB, maxc, fbuf, gs);
    }

    fm_argmax_kernel<<<tileGrid, blk128, 0, stream>>>(pA, pB, gs, maxc,
                                                      fbuf, rowmax, rowidx);
    fm_loss_kernel<<<(B_ * M_ + 255) / 256, 256, 0, stream>>>(x0, xgt, vpred, alpha,
                                                              rowmax, rowidx, acc);
    fm_finalize_kernel<<<1, 1, 0, stream>>>(acc, out);
}